// GLE_TS_11407433138448
// MI455X (gfx1250) — compile-verified
//
#include <hip/hip_runtime.h>

typedef __attribute__((ext_vector_type(2))) float v2f;
typedef __attribute__((ext_vector_type(8))) float v8f;

#define FL 100              // filter length L
#define K2 200              // fused dot length (2L)
#define WAVES 4
#define ROWS_PER_WAVE 16
#define ROWS_PER_BLOCK (WAVES * ROWS_PER_WAVE)
#define N3 600000           // NUM_ATOMS * 3

// ---------------------------------------------------------------------------
// Setup kernel: build the fused 200-float kernel ck in d_ws.
//   ck[j]      = -memory_kernel[99-j]   for j in [0,100)
//   ck[100+j'] =  h[99-j']              for j' in [0,100)
// memory_kernel[k] = racf[k] * mass[0][0] * dt[0] / (BOLTZMAN * T)
// racf[k] = sum_{j=0}^{99-k} h[k+j]*h[j]
// ---------------------------------------------------------------------------
__global__ void gle_ck_kernel(const float* __restrict__ h,
                              const float* __restrict__ mass,
                              const float* __restrict__ dt,
                              const int* __restrict__ T,
                              float* __restrict__ ck) {
  int k = threadIdx.x;
  if (k < FL) {
    float acc = 0.f;
    for (int j = 0; j + k < FL; ++j) acc += h[k + j] * h[j];
    float mk = acc * mass[0] * dt[0] / (0.001987191f * (float)T[0]);
    ck[(FL - 1) - k] = -mk;
  } else if (k < 2 * FL) {
    ck[k] = h[2 * FL - 1 - k];
  }
}

// ---------------------------------------------------------------------------
// Main kernel: each wave computes 16 rows of out[i] = dot(rowdata_i, ck)
// via 50 x V_WMMA_F32_16X16X4_F32 (B = ck broadcast across all 16 columns,
// so every column of D equals the dot product).
// ---------------------------------------------------------------------------
__global__ __launch_bounds__(WAVES * 32) void gle_main_kernel(
    const float* __restrict__ v,
    const float* __restrict__ v_list,
    const float* __restrict__ w_list,
    const float* __restrict__ w_new,
    const float* __restrict__ ck,
    float* __restrict__ out) {
  __shared__ float ck_s[K2];
  __shared__ float tile[WAVES * ROWS_PER_WAVE * K2];  // 51.2 KB

  const int tid = threadIdx.x;
  const int lane = tid & 31;
  const int wave = tid >> 5;
  const int rowBase = blockIdx.x * ROWS_PER_BLOCK + wave * ROWS_PER_WAVE;
  float* myTile = &tile[wave * ROWS_PER_WAVE * K2];

  if (tid < K2) ck_s[tid] = ck[tid];

  // Stage this wave's 16x200 f32 tile into LDS with coalesced b32 loads.
  // Row layout: [v_list[row,102..200], v[row], w_list[row,102..200], w_new[row]]
  for (int r = 0; r < ROWS_PER_WAVE; ++r) {
    const size_t grow = (size_t)(rowBase + r);
    const float* __restrict__ vrow = v_list + grow * 201;
    const float* __restrict__ wrow = w_list + grow * 201;
    float* __restrict__ trow = myTile + r * K2;
    // v-half: k in [0,100): vrow[102+k] for k<99, else v[grow]
    for (int k = lane; k < FL; k += 32) {
      const float* p = (k < FL - 1) ? (vrow + 102 + k) : (v + grow);
      trow[k] = *p;
    }
    // w-half: k in [0,100): wrow[102+k] for k<99, else w_new[grow]
    for (int k = lane; k < FL; k += 32) {
      const float* p = (k < FL - 1) ? (wrow + 102 + k) : (w_new + grow);
      trow[FL + k] = *p;
    }
  }
  __syncthreads();

  // WMMA accumulate: A lane mapping (16x4 f32): row = lane&15,
  // K = kb + 2*(lane>=16) + {0,1}. B uses the same per-lane K mapping with
  // ck broadcast to every column.
  const int row = lane & 15;
  const int khalf = (lane >> 4) << 1;  // 0 or 2
  const float* arow = &myTile[row * K2];
  v8f c = {};
#pragma unroll 10
  for (int kb = 0; kb < K2; kb += 4) {
    const int K = kb + khalf;
    v2f a = *(const v2f*)(arow + K);   // 8B-aligned ds_load_b64
    v2f b = *(const v2f*)(ck_s + K);
    c = __builtin_amdgcn_wmma_f32_16x16x4_f32(
        /*neg_a=*/false, a, /*neg_b=*/false, b,
        /*c_mod=*/(short)0, c, /*reuse_a=*/false, /*reuse_b=*/false);
  }

  // D layout (32-bit C/D 16x16): lanes 0-15 hold rows 0-7 in VGPR 0-7,
  // lanes 16-31 hold rows 8-15. All columns identical -> any lane valid.
  // Branchless 3-level select tree (7 v_cndmask) instead of a branch ladder.
  const int r8 = lane & 7;
  const bool b0 = (r8 & 1) != 0;
  const bool b1 = (r8 & 2) != 0;
  const bool b2 = (r8 & 4) != 0;
  float s0 = b0 ? c[1] : c[0];
  float s1 = b0 ? c[3] : c[2];
  float s2 = b0 ? c[5] : c[4];
  float s3 = b0 ? c[7] : c[6];
  float t0 = b1 ? s1 : s0;
  float t1 = b1 ? s3 : s2;
  float res = b2 ? t1 : t0;
  if ((lane & 15) < 8) {
    out[rowBase + (lane >> 4) * 8 + r8] = res;  // coalesced 8-lane b32 stores
  }
}

extern "C" void kernel_launch(void* const* d_in, const int* in_sizes, int n_in,
                              void* d_out, int out_size, void* d_ws, size_t ws_size,
                              hipStream_t stream) {
  (void)in_sizes; (void)n_in; (void)out_size; (void)ws_size;
  const float* h      = (const float*)d_in[0];
  const float* v      = (const float*)d_in[1];
  const float* v_list = (const float*)d_in[2];
  const float* w_list = (const float*)d_in[3];
  const float* w_new  = (const float*)d_in[4];
  const float* mass   = (const float*)d_in[5];
  const float* dt     = (const float*)d_in[6];
  const int*   T      = (const int*)d_in[7];

  float* ck  = (float*)d_ws;   // 200 floats of scratch
  float* out = (float*)d_out;

  gle_ck_kernel<<<1, 256, 0, stream>>>(h, mass, dt, T, ck);

  const int blocks = N3 / ROWS_PER_BLOCK;  // 600000 / 64 = 9375 exactly
  gle_main_kernel<<<blocks, WAVES * 32, 0, stream>>>(v, v_list, w_list, w_new,
                                                     ck, out);
}